// SSMLayer_33767032882062
// MI455X (gfx1250) — compile-verified
//
#include <hip/hip_runtime.h>
#include <hip/hip_bf16.h>
#include <math.h>

typedef __attribute__((ext_vector_type(2))) float v2f;
typedef __attribute__((ext_vector_type(8))) float v8f;

#define DIM_    1024
#define DINNER  2048
#define NSTATE  16
#define DTRANK  64
#define DCONV   4
#define BATCH_  2
#define SEQ_    1024
#define MROWS   (BATCH_*SEQ_)       // 2048
#define NCHUNK  16
#define CHUNK   (SEQ_/NCHUNK)       // 64
#define XDBLW   (DTRANK + 2*NSTATE) // 96

// ---------------------------------------------------------------------------
// Generic fp32 WMMA GEMM: C[M,N] = A[M,K] @ B[K,N], row-major, 16|M, 4|K.
// Block = 4 waves stacked in M; each wave computes a 16 x (16*NT) strip with
// V_WMMA_F32_16X16X4_F32, reusing the A fragment across NT column tiles.
// EPI==1: C = softplus(acc + bias[n])  (dt_proj epilogue)
// f32 16x16x4 fragment layout (ISA 7.12.2):
//   A(16x4):  lane = m + 16*(k>>1), vgpr = k&1   -> per-lane b64 load
//   B(4x16):  lane = n + 16*(k>>1), vgpr = k&1   -> two coalesced b32 loads
//   C/D:      lane = n + 16*(m>>3), vgpr = m&7
// ---------------------------------------------------------------------------
template<int NT, int EPI>
__global__ __launch_bounds__(128)
void gemm_wmma_f32(const float* __restrict__ A, int lda,
                   const float* __restrict__ B, int ldb,
                   float* __restrict__ C, int ldc,
                   const float* __restrict__ bias, int K)
{
    const int lane = threadIdx.x & 31;
    const int wave = threadIdx.x >> 5;
    const int lhi  = lane >> 4;       // 0 or 1
    const int llo  = lane & 15;
    const int row0 = (blockIdx.y * 4 + wave) * 16;
    const int col0 = blockIdx.x * (16 * NT);

    v8f acc[NT] = {};

    const float* Ap = A + (size_t)(row0 + llo) * lda + 2 * lhi;
    const float* Bp = B + (size_t)(2 * lhi) * ldb + col0 + llo;

    for (int k0 = 0; k0 < K; k0 += 4) {
        v2f a = *(const v2f*)(Ap + k0);
#pragma unroll
        for (int t = 0; t < NT; ++t) {
            v2f b;
            b.x = Bp[(size_t)k0 * ldb + t * 16];
            b.y = Bp[(size_t)(k0 + 1) * ldb + t * 16];
            acc[t] = __builtin_amdgcn_wmma_f32_16x16x4_f32(
                false, a, false, b, (short)0, acc[t], false, false);
        }
    }

    float* Cp = C + (size_t)(row0 + 8 * lhi) * ldc + col0 + llo;
#pragma unroll
    for (int t = 0; t < NT; ++t) {
        float bval = 0.f;
        if (EPI == 1) bval = bias[col0 + t * 16 + llo];
#pragma unroll
        for (int i = 0; i < 8; ++i) {
            float v = acc[t][i];
            if (EPI == 1) {
                v += bval;
                v = (v > 20.f) ? v : log1pf(__expf(v));   // softplus
            }
            Cp[(size_t)i * ldc + t * 16] = v;
        }
    }
}

// ---------------------------------------------------------------------------
// Depthwise causal conv (k=4) + bias + SiLU.  h[b,t,d]
// ---------------------------------------------------------------------------
__global__ __launch_bounds__(256)
void conv_silu_kernel(const float* __restrict__ xz, const float* __restrict__ cw,
                      const float* __restrict__ cb, float* __restrict__ h)
{
    int idx = blockIdx.x * 256 + threadIdx.x;        // B*SEQ*DINNER = 2^22
    int d = idx & (DINNER - 1);
    int t = (idx >> 11) & (SEQ_ - 1);
    int b = idx >> 21;
    const float* xp = xz + ((size_t)b * SEQ_) * (2 * DINNER) + d;  // row stride 4096
    float acc = cb[d];
#pragma unroll
    for (int j = 0; j < DCONV; ++j) {
        int tt = t - (DCONV - 1) + j;
        if (tt >= 0) acc += xp[(size_t)tt * (2 * DINNER)] * cw[d * DCONV + j];
    }
    h[idx] = acc / (1.f + __expf(-acc));             // x*sigmoid(x)
}

// ---------------------------------------------------------------------------
// Scan pass 1: per (b,d,chunk) compute chunk-local state from zero init and
// sum of deltas (chunk decay = exp(A[n]*sum_delta)).  65536 threads, depth 64.
// ---------------------------------------------------------------------------
__global__ __launch_bounds__(256)
void scan_pass1(const float* __restrict__ delta, const float* __restrict__ h,
                const float* __restrict__ xdbl, const float* __restrict__ alog,
                float* __restrict__ cstate, float* __restrict__ sumd)
{
    int tid = blockIdx.x * 256 + threadIdx.x;        // B*NCHUNK*DINNER = 65536
    int d = tid & (DINNER - 1);
    int c = (tid >> 11) & (NCHUNK - 1);
    int b = tid >> 15;

    float a[NSTATE];
#pragma unroll
    for (int n = 0; n < NSTATE; ++n) a[n] = -__expf(alog[d * NSTATE + n]);

    float s[NSTATE] = {};
    float sd = 0.f;
    int t0 = c * CHUNK;
    for (int t = t0; t < t0 + CHUNK; ++t) {
        size_t m = (size_t)b * SEQ_ + t;
        float dl = delta[m * DINNER + d];
        float hh = h[m * DINNER + d];
        float du = dl * hh;
        const float* Bp = xdbl + m * XDBLW + DTRANK;
        float4 B0 = *(const float4*)(Bp + 0);
        float4 B1 = *(const float4*)(Bp + 4);
        float4 B2 = *(const float4*)(Bp + 8);
        float4 B3 = *(const float4*)(Bp + 12);
        float Bv[NSTATE] = {B0.x,B0.y,B0.z,B0.w, B1.x,B1.y,B1.z,B1.w,
                            B2.x,B2.y,B2.z,B2.w, B3.x,B3.y,B3.z,B3.w};
#pragma unroll
        for (int n = 0; n < NSTATE; ++n)
            s[n] = __expf(dl * a[n]) * s[n] + du * Bv[n];
        sd += dl;
    }
    size_t base = ((size_t)(b * DINNER + d)) * NCHUNK + c;
    sumd[base] = sd;
    float* cs = cstate + base * NSTATE;
#pragma unroll
    for (int n = 0; n < NSTATE; ++n) cs[n] = s[n];
}

// ---------------------------------------------------------------------------
// Scan pass 2: carry scan across the 16 chunks per (b,d).  4096 threads.
// carry_in[c] stored; carry = exp(A*sumd[c])*carry + cstate[c]
// ---------------------------------------------------------------------------
__global__ __launch_bounds__(256)
void scan_pass2(const float* __restrict__ alog, const float* __restrict__ sumd,
                const float* __restrict__ cstate, float* __restrict__ carry)
{
    int tid = blockIdx.x * 256 + threadIdx.x;        // B*DINNER = 4096
    int d = tid & (DINNER - 1);
    float a[NSTATE];
#pragma unroll
    for (int n = 0; n < NSTATE; ++n) a[n] = -__expf(alog[d * NSTATE + n]);

    float cr[NSTATE] = {};
    size_t base = (size_t)tid * NCHUNK;              // tid == b*DINNER + d
    for (int c = 0; c < NCHUNK; ++c) {
        float* out      = carry  + (base + c) * NSTATE;
        const float* cs = cstate + (base + c) * NSTATE;
        float sd = sumd[base + c];
#pragma unroll
        for (int n = 0; n < NSTATE; ++n) {
            out[n] = cr[n];
            cr[n] = __expf(sd * a[n]) * cr[n] + cs[n];
        }
    }
}

// ---------------------------------------------------------------------------
// Scan pass 3: replay each chunk from its carry-in, emit
// y = (state.C + h*D) * silu(res).  65536 threads, depth 64.
// ---------------------------------------------------------------------------
__global__ __launch_bounds__(256)
void scan_pass3(const float* __restrict__ delta, const float* __restrict__ h,
                const float* __restrict__ xdbl, const float* __restrict__ alog,
                const float* __restrict__ carry, const float* __restrict__ Dp,
                const float* __restrict__ xz, float* __restrict__ y)
{
    int tid = blockIdx.x * 256 + threadIdx.x;
    int d = tid & (DINNER - 1);
    int c = (tid >> 11) & (NCHUNK - 1);
    int b = tid >> 15;

    float a[NSTATE];
#pragma unroll
    for (int n = 0; n < NSTATE; ++n) a[n] = -__expf(alog[d * NSTATE + n]);

    float s[NSTATE];
    const float* cs = carry + (((size_t)(b * DINNER + d)) * NCHUNK + c) * NSTATE;
#pragma unroll
    for (int n = 0; n < NSTATE; ++n) s[n] = cs[n];

    float Dd = Dp[d];
    int t0 = c * CHUNK;
    for (int t = t0; t < t0 + CHUNK; ++t) {
        size_t m = (size_t)b * SEQ_ + t;
        float dl = delta[m * DINNER + d];
        float hh = h[m * DINNER + d];
        float du = dl * hh;
        const float* Bp = xdbl + m * XDBLW + DTRANK;
        float4 B0 = *(const float4*)(Bp + 0);
        float4 B1 = *(const float4*)(Bp + 4);
        float4 B2 = *(const float4*)(Bp + 8);
        float4 B3 = *(const float4*)(Bp + 12);
        float4 C0 = *(const float4*)(Bp + 16);
        float4 C1 = *(const float4*)(Bp + 20);
        float4 C2 = *(const float4*)(Bp + 24);
        float4 C3 = *(const float4*)(Bp + 28);
        float Bv[NSTATE] = {B0.x,B0.y,B0.z,B0.w, B1.x,B1.y,B1.z,B1.w,
                            B2.x,B2.y,B2.z,B2.w, B3.x,B3.y,B3.z,B3.w};
        float Cv[NSTATE] = {C0.x,C0.y,C0.z,C0.w, C1.x,C1.y,C1.z,C1.w,
                            C2.x,C2.y,C2.z,C2.w, C3.x,C3.y,C3.z,C3.w};
        float acc = 0.f;
#pragma unroll
        for (int n = 0; n < NSTATE; ++n) {
            s[n] = __expf(dl * a[n]) * s[n] + du * Bv[n];
            acc += s[n] * Cv[n];
        }
        float r = xz[m * (2 * DINNER) + DINNER + d];
        float sil = r / (1.f + __expf(-r));
        y[m * DINNER + d] = (acc + hh * Dd) * sil;
    }
}

// ---------------------------------------------------------------------------
extern "C" void kernel_launch(void* const* d_in, const int* in_sizes, int n_in,
                              void* d_out, int out_size, void* d_ws, size_t ws_size,
                              hipStream_t stream)
{
    (void)in_sizes; (void)n_in; (void)out_size; (void)ws_size;
    const float* x    = (const float*)d_in[0];
    const float* inw  = (const float*)d_in[1];
    const float* cw   = (const float*)d_in[2];
    const float* cb   = (const float*)d_in[3];
    const float* xpw  = (const float*)d_in[4];
    const float* dtw  = (const float*)d_in[5];
    const float* dtb  = (const float*)d_in[6];
    const float* alog = (const float*)d_in[7];
    const float* Dp   = (const float*)d_in[8];
    const float* outw = (const float*)d_in[9];
    float* out = (float*)d_out;

    // workspace carve-up (floats)
    float* ws    = (float*)d_ws;
    float* xz    = ws;                                          // 2048*4096
    float* hbuf  = xz    + (size_t)MROWS * 2 * DINNER;          // 2048*2048
    float* xdbl  = hbuf  + (size_t)MROWS * DINNER;              // 2048*96
    float* delta = xdbl  + (size_t)MROWS * XDBLW;               // 2048*2048
    float* sumd  = delta + (size_t)MROWS * DINNER;              // 2*2048*16
    float* cst   = sumd  + (size_t)BATCH_ * DINNER * NCHUNK;    // *16
    float* carry = cst   + (size_t)BATCH_ * DINNER * NCHUNK * NSTATE;
    float* ybuf  = carry + (size_t)BATCH_ * DINNER * NCHUNK * NSTATE; // 2048*2048

    dim3 blk(128);

    // 1) xz = x @ in_proj_w        (2048 x 1024 x 4096)
    gemm_wmma_f32<4,0><<<dim3((2*DINNER)/64, MROWS/64), blk, 0, stream>>>(
        x, DIM_, inw, 2*DINNER, xz, 2*DINNER, nullptr, DIM_);

    // 2) h = silu(depthwise_conv(xp) + b)
    conv_silu_kernel<<<(MROWS*DINNER)/256, 256, 0, stream>>>(xz, cw, cb, hbuf);

    // 3) x_dbl = h @ x_proj_w      (2048 x 2048 x 96)
    gemm_wmma_f32<3,0><<<dim3(XDBLW/48, MROWS/64), blk, 0, stream>>>(
        hbuf, DINNER, xpw, XDBLW, xdbl, XDBLW, nullptr, DINNER);

    // 4) delta = softplus(dlt @ dt_proj_w + b)   (2048 x 64 x 2048)
    gemm_wmma_f32<4,1><<<dim3(DINNER/64, MROWS/64), blk, 0, stream>>>(
        xdbl, XDBLW, dtw, DINNER, delta, DINNER, dtb, DTRANK);

    // 5-7) chunked selective scan
    scan_pass1<<<(BATCH_*DINNER*NCHUNK)/256, 256, 0, stream>>>(
        delta, hbuf, xdbl, alog, cst, sumd);
    scan_pass2<<<(BATCH_*DINNER)/256, 256, 0, stream>>>(alog, sumd, cst, carry);
    scan_pass3<<<(BATCH_*DINNER*NCHUNK)/256, 256, 0, stream>>>(
        delta, hbuf, xdbl, alog, carry, Dp, xz, ybuf);

    // 8) out = y @ out_proj_w      (2048 x 2048 x 1024)
    gemm_wmma_f32<4,0><<<dim3(DIM_/64, MROWS/64), blk, 0, stream>>>(
        ybuf, DINNER, outw, DIM_, out, DIM_, nullptr, DINNER);
}